// SmoothedComplex_46213848105402
// MI455X (gfx1250) — compile-verified
//
#include <hip/hip_runtime.h>
#include <math.h>
#include <stdint.h>

// ---------------------------------------------------------------------------
// SmoothedComplex loss on MI455X (gfx1250)
//
//   box64x64(pred), box64x64(tgt)  ->  atan2 per pixel  ->  mean |diff|
//
//  * /4096 kernel normalization cancels in atan2 -> box SUMS only.
//  * Vertical 64-row sliding sum: 88-slot LDS ring fed by
//    global_load_async_to_lds_b32 (ASYNCcnt, 24-row prefetch lead to cover
//    HBM latency at 1 WG/WGP occupancy). Each element read from HBM once
//    per strip; adjacent-strip halo overlap is absorbed by the 192MB L2.
//  * Horizontal 64-tap sum: three 16x16 f32 matmuls with constant
//    triangular/ones matrices via chained V_WMMA_F32_16X16X4_F32.
//    THREE independent accumulator chains (depth 4 each) instead of one
//    depth-12 chain -> matrix-pipe latency hiding; merged with v_pk_add.
//  * Deterministic: per-WG partials -> d_ws, fixed-order finalize kernel.
// ---------------------------------------------------------------------------

#define NMASK   64
#define W       1024
#define H       1024
#define OUTW    961          // 1024 - 63
#define STRIPW  64           // output columns per workgroup
#define INW     128          // input columns needed: STRIPW + 63 (padded)
#define RING    88           // > 63 + LEAD  (overwrite-after-last-use safe)
#define LEAD    24           // async prefetch lead rows (48 outstanding <= 63)
#define BUFP    132          // colsum buffer row pitch (bank-conflict pad)
#define BOXP    68           // box buffer row pitch (bank-conflict pad)

typedef float v2f __attribute__((ext_vector_type(2)));
typedef float v8f __attribute__((ext_vector_type(8)));

__device__ __forceinline__ void async_f32_to_lds(const float* g, unsigned lds_off) {
    asm volatile("global_load_async_to_lds_b32 %0, %1, off"
                 :: "v"(lds_off), "v"((unsigned long long)(uintptr_t)g)
                 : "memory");
}
__device__ __forceinline__ void wait_async_row_landed() {
    // <=46 outstanding == rows y+1..y+23 (2 instr/row) in flight -> row y landed
    asm volatile("s_wait_asynccnt 46" ::: "memory");
}

__global__ __launch_bounds__(256) void smooth_angle_l1(
    const float* __restrict__ pred, const float* __restrict__ tgt,
    float* __restrict__ ws)
{
    __shared__ float ring[4][RING][INW];   // 180224 B  input-row history
    __shared__ float cbuf[4][16][BUFP];    //  33792 B  16 rows of column sums
    __shared__ float bbuf[4][16][BOXP];    //  17408 B  16x64 box outputs
    __shared__ float red[256];

    const int tid   = threadIdx.x;
    const int wg    = blockIdx.x;
    const int batch = wg >> 4;
    const int strip = wg & 15;
    const int x0    = strip * STRIPW;

    const int pp   = tid >> 7;                 // 0 = pred planes, 1 = tgt planes
    const int xi   = tid & 127;                // column within strip halo
    const int colg = min(x0 + xi, W - 1);      // clamp: junk adds cancel in sliding sum

    const float* src   = pp ? tgt : pred;
    const float* baseA = src + ((size_t)batch * 2 + 0) * (size_t)(W * H) + colg;
    const float* baseB = src + ((size_t)batch * 2 + 1) * (size_t)(W * H) + colg;
    const int cA = 2 * pp, cB = 2 * pp + 1;

    const int lane = tid & 31;
    const int wave = tid >> 5;

    // prologue: prefetch rows 0..LEAD-1 into the ring
    for (int r = 0; r < LEAD; ++r) {
        async_f32_to_lds(baseA + (size_t)r * W, (unsigned)(uintptr_t)&ring[cA][r][xi]);
        async_f32_to_lds(baseB + (size_t)r * W, (unsigned)(uintptr_t)&ring[cB][r][xi]);
    }

    float accA = 0.f, accB = 0.f;   // running 64-row column sums (this thread's column)
    float part = 0.f;               // |delta angle| partial sum

#pragma unroll 1
    for (int y = 0; y < H; ++y) {
        wait_async_row_landed();
        const int sl = y % RING;
        accA += ring[cA][sl][xi];
        accB += ring[cB][sl][xi];

        if (y >= NMASK - 1) {
            const int yo = y - (NMASK - 1);     // output row
            const int r  = yo & 15;
            cbuf[cA][r][xi] = accA;
            cbuf[cB][r][xi] = accB;

            if (r == 15 || yo == OUTW - 1) {
                // ---- flush a 16-row group: WMMA horizontal box + atan2 ----
                __syncthreads();
                const int g = yo >> 4;
                // 16 tile jobs: (plane 0..3) x (jt 0,16,32,48); 8 waves x 2
#pragma unroll
                for (int i = 0; i < 2; ++i) {
                    const int job  = wave + 8 * i;
                    const int p    = job >> 2;
                    const int jt   = (job & 3) << 4;
                    const int half = lane >> 4;
                    const int m    = lane & 15;
                    v8f c0 = {}, c1 = {}, c2 = {};   // 3 independent chains
#pragma unroll
                    for (int kk = 0; kk < 4; ++kk) {
                        const int k0 = 4 * kk + (half ? 2 : 0);
                        const int k1 = k0 + 1;
                        v2f a, b;
                        // S0 x M0, M0[k][n] = (k >= n): head partial window
                        a.x = cbuf[p][m][jt + k0];
                        a.y = cbuf[p][m][jt + k1];
                        b.x = (k0 >= m) ? 1.f : 0.f;
                        b.y = (k1 >= m) ? 1.f : 0.f;
                        c0 = __builtin_amdgcn_wmma_f32_16x16x4_f32(false, a, false, b,
                                                                   (short)0, c0, false, false);
                        // (S1+S2+S3) x ONES: fully-covered middle chunks
                        a.x = cbuf[p][m][jt + 16 + k0] + cbuf[p][m][jt + 32 + k0] +
                              cbuf[p][m][jt + 48 + k0];
                        a.y = cbuf[p][m][jt + 16 + k1] + cbuf[p][m][jt + 32 + k1] +
                              cbuf[p][m][jt + 48 + k1];
                        b.x = 1.f; b.y = 1.f;
                        c1 = __builtin_amdgcn_wmma_f32_16x16x4_f32(false, a, false, b,
                                                                   (short)0, c1, false, false);
                        // S4 x M4, M4[k][n] = (k < n): tail partial window
                        a.x = cbuf[p][m][jt + 64 + k0];
                        a.y = cbuf[p][m][jt + 64 + k1];
                        b.x = (k0 < m) ? 1.f : 0.f;
                        b.y = (k1 < m) ? 1.f : 0.f;
                        c2 = __builtin_amdgcn_wmma_f32_16x16x4_f32(false, a, false, b,
                                                                   (short)0, c2, false, false);
                    }
#pragma unroll
                    for (int v = 0; v < 8; ++v)
                        bbuf[p][v + 8 * half][jt + m] = c0[v] + c1[v] + c2[v];
                }
                __syncthreads();
                // atan2 + |diff| over the 16x64 tile (VALU)
#pragma unroll
                for (int q = 0; q < 4; ++q) {
                    const int px = tid + q * 256;
                    const int r2 = px >> 6;
                    const int j  = px & 63;
                    const int gy = (g << 4) + r2;
                    const int gx = x0 + j;
                    if (gy < OUTW && gx < OUTW) {
                        const float pa = atan2f(bbuf[1][r2][j], bbuf[0][r2][j]);
                        const float ta = atan2f(bbuf[3][r2][j], bbuf[2][r2][j]);
                        part += fabsf(pa - ta);
                    }
                }
            }
            // slide window: drop row yo (its ring slot is overwritten >=1 iter later)
            const int so = yo % RING;
            accA -= ring[cA][so][xi];
            accB -= ring[cB][so][xi];
        }

        // prefetch row y + LEAD
        const int yn = y + LEAD;
        if (yn < H) {
            const int sn = yn % RING;
            async_f32_to_lds(baseA + (size_t)yn * W, (unsigned)(uintptr_t)&ring[cA][sn][xi]);
            async_f32_to_lds(baseB + (size_t)yn * W, (unsigned)(uintptr_t)&ring[cB][sn][xi]);
        }
    }

    // deterministic block reduction
    red[tid] = part;
    __syncthreads();
    for (int s = 128; s > 0; s >>= 1) {
        if (tid < s) red[tid] += red[tid + s];
        __syncthreads();
    }
    if (tid == 0) ws[wg] = red[0];
}

__global__ __launch_bounds__(256) void finalize_mean(
    const float* __restrict__ ws, float* __restrict__ out)
{
    __shared__ float red[256];
    const int tid = threadIdx.x;
    red[tid] = ws[tid];
    __syncthreads();
    for (int s = 128; s > 0; s >>= 1) {
        if (tid < s) red[tid] += red[tid + s];
        __syncthreads();
    }
    if (tid == 0)
        out[0] = red[0] / (16.0f * 961.0f * 961.0f);
}

extern "C" void kernel_launch(void* const* d_in, const int* in_sizes, int n_in,
                              void* d_out, int out_size, void* d_ws, size_t ws_size,
                              hipStream_t stream) {
    (void)in_sizes; (void)n_in; (void)out_size; (void)ws_size;
    const float* pred = (const float*)d_in[0];
    const float* tgt  = (const float*)d_in[1];
    float* ws = (float*)d_ws;          // 256 floats of scratch
    smooth_angle_l1<<<dim3(256), dim3(256), 0, stream>>>(pred, tgt, ws);
    finalize_mean<<<dim3(1), dim3(256), 0, stream>>>(ws, (float*)d_out);
}